// MinibatchDiscrimination_3504693314169
// MI455X (gfx1250) — compile-verified
//
#include <hip/hip_runtime.h>

typedef __attribute__((ext_vector_type(2))) float v2f;
typedef __attribute__((ext_vector_type(8))) float v8f;

#define B_SZ 512
#define F_SZ 1024
#define O_SZ 1024   // NUM_KERNELS * KERNEL_DIM
#define NK   64
#define KD   16

// ---------------------------------------------------------------------------
// Stage 1: M = x @ T via V_WMMA_F32_16X16X4_F32 (exact f32 matrix path).
// 2048 tiles of 16x16 (32 M-tiles x 64 N-tiles); 1 tile per wave, 8 waves/blk.
// ---------------------------------------------------------------------------
__global__ __launch_bounds__(256) void mbd_gemm_wmma_f32(
    const float* __restrict__ x,   // [512, 1024] row-major
    const float* __restrict__ T,   // [1024, 1024] row-major
    float* __restrict__ M)         // [512, 1024] row-major
{
  const int lane = threadIdx.x & 31;
  const int wave = threadIdx.x >> 5;
  const int tile = blockIdx.x * 8 + wave;      // 0..2047
  const int tm = (tile & 31) << 4;             // row tile base (B dim)
  const int tn = (tile >> 5) << 4;             // col tile base (O dim)

  const int half = lane >> 4;                  // 0: lanes 0-15, 1: lanes 16-31
  const int l15  = lane & 15;

  const float* xrow = x + (tm + l15) * F_SZ;   // A: this lane's matrix row
  const float* tcol = T + tn + l15;            // B: this lane's matrix col

  v8f acc = {};
  for (int k = 0; k < F_SZ; k += 4) {
    // A fragment 16x4: half 0 -> (K=k,k+1), half 1 -> (K=k+2,k+3)
    v2f a = *(const v2f*)(xrow + k + half * 2);
    // B fragment 4x16: half 0 -> rows k,k+1 ; half 1 -> rows k+2,k+3
    v2f b;
    b.x = tcol[(k + half * 2) * O_SZ];
    b.y = tcol[(k + half * 2 + 1) * O_SZ];
    acc = __builtin_amdgcn_wmma_f32_16x16x4_f32(
        /*neg_a=*/false, a, /*neg_b=*/false, b,
        /*c_mod=*/(short)0, acc, /*reuse_a=*/false, /*reuse_b=*/false);
  }

  // C/D layout: VGPR r -> row (tm + r + half*8), col (tn + l15)
  #pragma unroll
  for (int r = 0; r < 8; ++r) {
    M[(tm + r + half * 8) * O_SZ + tn + l15] = acc[r];
  }
}

// ---------------------------------------------------------------------------
// Stage 2: out[i,k] = sum_j exp(-sum_d |M[j,k,d]-M[i,k,d]|) - 1
// Block = (i-half, k). 512x16 f32 slice for kernel k staged in 32KB LDS;
// j-sweep reads are wave-uniform -> LDS broadcast.
// ---------------------------------------------------------------------------
__global__ __launch_bounds__(256) void mbd_pairwise_l1_exp(
    const float* __restrict__ M,   // [512, 1024]
    float* __restrict__ out)       // [512, 64]
{
  __shared__ float4 sh[B_SZ * 4];  // 512 rows x 16 floats = 32 KB

  const int k     = blockIdx.y;    // 0..63
  const int itile = blockIdx.x;    // 0..1
  const int tid   = threadIdx.x;   // 0..255

  // Cooperative load: rows [0,512) x 16 floats of kernel-k slice.
  const float4* M4 = (const float4*)M;     // row stride = 256 float4
  for (int t = tid; t < B_SZ * 4; t += 256) {
    const int row = t >> 2;
    const int q   = t & 3;
    sh[t] = M4[row * (O_SZ / 4) + k * 4 + q];
  }
  __syncthreads();

  const int i = itile * 256 + tid;
  const float4 a0 = sh[i * 4 + 0];
  const float4 a1 = sh[i * 4 + 1];
  const float4 a2 = sh[i * 4 + 2];
  const float4 a3 = sh[i * 4 + 3];

  float acc = 0.0f;
  #pragma unroll 4
  for (int j = 0; j < B_SZ; ++j) {
    const float4 b0 = sh[j * 4 + 0];
    const float4 b1 = sh[j * 4 + 1];
    const float4 b2 = sh[j * 4 + 2];
    const float4 b3 = sh[j * 4 + 3];
    float l1 = fabsf(a0.x - b0.x) + fabsf(a0.y - b0.y)
             + fabsf(a0.z - b0.z) + fabsf(a0.w - b0.w)
             + fabsf(a1.x - b1.x) + fabsf(a1.y - b1.y)
             + fabsf(a1.z - b1.z) + fabsf(a1.w - b1.w)
             + fabsf(a2.x - b2.x) + fabsf(a2.y - b2.y)
             + fabsf(a2.z - b2.z) + fabsf(a2.w - b2.w)
             + fabsf(a3.x - b3.x) + fabsf(a3.y - b3.y)
             + fabsf(a3.z - b3.z) + fabsf(a3.w - b3.w);
    acc += __expf(-l1);
  }

  out[i * NK + k] = acc - 1.0f;   // remove self term exp(0)
}

extern "C" void kernel_launch(void* const* d_in, const int* in_sizes, int n_in,
                              void* d_out, int out_size, void* d_ws, size_t ws_size,
                              hipStream_t stream) {
  const float* x = (const float*)d_in[0];   // [512, 1024]
  const float* T = (const float*)d_in[1];   // [1024, 1024]
  float* out = (float*)d_out;               // [512, 64]
  float* M   = (float*)d_ws;                // [512, 1024] scratch (2 MB)

  // Stage 1: GEMM on the WMMA units (2048 tiles / 8 waves per block).
  mbd_gemm_wmma_f32<<<dim3(256), dim3(256), 0, stream>>>(x, T, M);

  // Stage 2: pairwise L1 + exp + row-sum.
  mbd_pairwise_l1_exp<<<dim3(2, 64), dim3(256), 0, stream>>>(M, out);
}